// BabyRadixAttention_41678362640762
// MI455X (gfx1250) — compile-verified
//
#include <hip/hip_runtime.h>
#include <math.h>
#include <stdint.h>

typedef __attribute__((ext_vector_type(2))) float v2f;
typedef __attribute__((ext_vector_type(8))) float v8f;

namespace {
constexpr int B  = 32;
constexpr int H  = 32;
constexpr int KH = 8;
constexpr int G  = H / KH;          // 4 q-heads per kv-head
constexpr int D  = 128;
constexpr int S  = 2049;
constexpr int NT = (S + 15) / 16;   // 129 key tiles
constexpr int SP = NT * 16;         // 2064 padded score length

// ---- LDS layout (in floats), total 15768 floats = 61.6 KB ----
constexpr int QS_PITCH = 132;       // bank-advance 4/row
constexpr int SC_PITCH = 2068;      // bank-advance 20/row
constexpr int KA_PITCH = 20;        // 80B/row: b128-aligned, 16 strided rows conflict-free
constexpr int KA_BUF   = 16 * KA_PITCH;        // 320 floats per K chunk buffer
constexpr int VS_PITCH = 132;       // 528B/row: b128-aligned
constexpr int VS_BUF   = 16 * VS_PITCH;        // 2112 floats per V tile buffer

constexpr int QS_OFF   = 0;                             // 16*132       = 2112
constexpr int SC_OFF   = QS_OFF + 16 * QS_PITCH;        // 4*2068       = 8272
constexpr int KST_OFF  = SC_OFF + 4 * SC_PITCH;         // 8*2*320      = 5120  (phase C: 2*2112 V overlay fits)
constexpr int RED_OFF  = KST_OFF + 8 * 2 * KA_BUF;      // 4*64         = 256
constexpr int STAT_OFF = RED_OFF + 256;                 // 8
constexpr int SMEM_FLOATS = STAT_OFF + 8;               // 15768 -> 63072 B
} // namespace

// ---- CDNA5 async global->LDS copy (ASYNCcnt-tracked DMA, no VGPR bounce) ----
__device__ __forceinline__ void async_copy_b128(unsigned lds_byte, unsigned long long gaddr) {
    asm volatile("global_load_async_to_lds_b128 %0, %1, off"
                 :: "v"(lds_byte), "v"(gaddr) : "memory");
}
// asm (not builtin) waits: the "memory" clobber makes them compiler-level
// fences, so LDS reads/writes cannot migrate across the hardware wait.
#define WAIT_ASYNC(n) asm volatile("s_wait_asynccnt %0" :: "i"(n) : "memory")
// Drain outstanding DS reads of a buffer before an async copy re-targets it
// (async LDS writes are unordered vs in-flight DS reads per the ISA; the
// round-3 disasm showed the scheduler sinking WMMAs below the next issue,
// voiding the implicit dscnt ordering).
#define DRAIN_DS()    asm volatile("s_wait_dscnt 0" ::: "memory")

__device__ __forceinline__ v8f wmma_f32(v2f a, v2f b, v8f c) {
    return __builtin_amdgcn_wmma_f32_16x16x4_f32(false, a, false, b, (short)0, c, false, false);
}

__global__ __launch_bounds__(256)
void decode_gqa_wmma_f32(const float* __restrict__ q,
                         const float* __restrict__ kn,
                         const float* __restrict__ vn,
                         const float* __restrict__ kbuf,
                         const float* __restrict__ vbuf,
                         const int*   __restrict__ req_to_token,
                         const int*   __restrict__ req_pool_indices,
                         const int*   __restrict__ out_cache_loc,
                         const int*   __restrict__ seq_lens,
                         float*       __restrict__ out)
{
    __shared__ __align__(16) float smem[SMEM_FLOATS];

    const int bkh = blockIdx.x;
    const int b   = bkh / KH;
    const int kh  = bkh % KH;
    const int tid = threadIdx.x;
    const int wv  = tid >> 5;        // wave id 0..7
    const int ln  = tid & 31;        // lane id
    const int lo  = ln & 15;
    const bool hiH = (ln >= 16);
    const int hoff = hiH ? 2 : 0;    // f32 WMMA frag: hi half holds K=+2,+3

    const int rpi = req_pool_indices[b];
    const int ocl = out_cache_loc[b];
    const int L   = seq_lens[b];
    const float scale = 0.08838834764831845f;  // 1/sqrt(128)

    const float* knew = kn + ((size_t)b * KH + kh) * D;
    const float* vnew = vn + ((size_t)b * KH + kh) * D;

    // low 32 bits of a flat shared pointer == LDS byte offset of our allocation
    const unsigned smem_lds = (unsigned)(uintptr_t)(void*)smem;

    // ---------------- stage Q: G real rows, zero-padded to 16 ----------------
    for (int j = tid; j < 16 * D; j += 256) {
        int row = j >> 7, col = j & (D - 1);
        float qv = 0.0f;
        if (row < G) qv = q[((size_t)b * H + (size_t)kh * G + row) * D + col];
        smem[QS_OFF + row * QS_PITCH + col] = qv;
    }
    __syncthreads();

    // ============ Phase A: scores = Q*K^T, per-wave async-pipelined ==========
    // Staging: chunk = 16 keys x 16 dims (64B/row). One async b128 covers 8
    // rows (lane l -> row l>>2, bytes (l&3)*16). Two asyncs per chunk,
    // double-buffered: issue chunk c+1, s_wait_asynccnt 2, compute chunk c.
    {
        const int rlo  = ln >> 2;            // this lane's rows: rlo, rlo+8
        const int cseg = (ln & 3) * 16;      // byte segment within 64B row-chunk
        const unsigned kw_lds = smem_lds + (unsigned)(KST_OFF + wv * 2 * KA_BUF) * 4u;

        for (int t = wv; t < NT; t += 8) {
            // per-lane row sources (new decode token -> kn input)
            const float* srcA; const float* srcB; bool zfA, zfB;
            {
                int s = t * 16 + rlo;
                if (s < S) {
                    int tok = req_to_token[(size_t)rpi * S + s];
                    srcA = (tok == ocl) ? knew : (kbuf + ((size_t)tok * KH + kh) * D);
                    zfA = false;
                } else { srcA = knew; zfA = true; }
                s += 8;
                if (s < S) {
                    int tok = req_to_token[(size_t)rpi * S + s];
                    srcB = (tok == ocl) ? knew : (kbuf + ((size_t)tok * KH + kh) * D);
                    zfB = false;
                } else { srcB = knew; zfB = true; }
            }
            auto issue = [&](int c, int p) {
                unsigned bb = kw_lds + (unsigned)(p * KA_BUF * 4);
                async_copy_b128(bb + (unsigned)(rlo * KA_PITCH * 4) + cseg,
                                (unsigned long long)(uintptr_t)srcA + (unsigned)(c * 64) + cseg);
                async_copy_b128(bb + (unsigned)((rlo + 8) * KA_PITCH * 4) + cseg,
                                (unsigned long long)(uintptr_t)srcB + (unsigned)(c * 64) + cseg);
            };

            v8f c = {};
            DRAIN_DS();                  // prior reads of buf0 (prev tile) done
            issue(0, 0);
            for (int ch = 0; ch < 8; ++ch) {
                if (ch < 7) {
                    DRAIN_DS();          // frag reads of target buffer drained
                    issue(ch + 1, (ch + 1) & 1);
                    WAIT_ASYNC(2);
                } else {
                    WAIT_ASYNC(0);
                }
                const int p = ch & 1;
                float* kb_ = &smem[KST_OFF + (wv * 2 + p) * KA_BUF];
                if (zfA) *(float4*)(kb_ + rlo * KA_PITCH + (ln & 3) * 4) =
                             make_float4(0.f, 0.f, 0.f, 0.f);
                if (zfB) *(float4*)(kb_ + (rlo + 8) * KA_PITCH + (ln & 3) * 4) =
                             make_float4(0.f, 0.f, 0.f, 0.f);
                for (int d0 = 0; d0 < 16; d0 += 4) {
                    v2f a  = *(const v2f*)(&smem[QS_OFF + lo * QS_PITCH + ch * 16 + d0 + hoff]);
                    v2f bb = *(const v2f*)(kb_ + lo * KA_PITCH + d0 + hoff);
                    c = wmma_f32(a, bb, c);
                }
            }
            // D layout: vgpr r = q-head g=r (r<4 valid), lanes 0-15 = key cols
            if (!hiH) {
                int s = t * 16 + lo;
                bool valid = (s < L);
                float* sp = &smem[SC_OFF + t * 16 + lo];
                sp[0 * SC_PITCH] = valid ? c[0] * scale : -3.0e38f;
                sp[1 * SC_PITCH] = valid ? c[1] * scale : -3.0e38f;
                sp[2 * SC_PITCH] = valid ? c[2] * scale : -3.0e38f;
                sp[3 * SC_PITCH] = valid ? c[3] * scale : -3.0e38f;
            }
        }
    }
    __syncthreads();

    // ---------------- Phase B: softmax over S, entirely in LDS ---------------
    {
        const int g  = tid >> 6;
        const int i0 = tid & 63;
        float* row = &smem[SC_OFF + g * SC_PITCH];
        float m = -3.0e38f;
        for (int i = i0; i < SP; i += 64) m = fmaxf(m, row[i]);
        smem[RED_OFF + g * 64 + i0] = m;
        __syncthreads();
        if (i0 == 0) {
            float mm = -3.0e38f;
            for (int i = 0; i < 64; ++i) mm = fmaxf(mm, smem[RED_OFF + g * 64 + i]);
            smem[STAT_OFF + g] = mm;
        }
        __syncthreads();
        const float rm = smem[STAT_OFF + g];
        float ss = 0.f;
        for (int i = i0; i < SP; i += 64) {
            float p = __expf(row[i] - rm);
            row[i] = p;
            ss += p;
        }
        smem[RED_OFF + g * 64 + i0] = ss;
        __syncthreads();
        if (i0 == 0) {
            float st = 0.f;
            for (int i = 0; i < 64; ++i) st += smem[RED_OFF + g * 64 + i];
            smem[STAT_OFF + 4 + g] = 1.0f / st;
        }
        __syncthreads();
    }

    // ============ Phase C: out = P*V, cooperative async double-buffer ========
    // Each wave async-copies its 2 rows of tile t+1 (1 async b128 per row:
    // lane l -> bytes l*16), waits its own ASYNCcnt, barrier, computes tile t.
    v8f acc = {};
    {
        const int r0 = 2 * wv, r1 = 2 * wv + 1;
        auto vsrc = [&](int s) -> const float* {
            if (s >= S) return vnew;          // p==0 there; any finite data is fine
            int tok = req_to_token[(size_t)rpi * S + s];
            return (tok == ocl) ? vnew : (vbuf + ((size_t)tok * KH + kh) * D);
        };
        auto issueV = [&](int t, int p) {
            unsigned bb = smem_lds + (unsigned)(KST_OFF + p * VS_BUF) * 4u;
            const float* sA = vsrc(t * 16 + r0);
            const float* sB = vsrc(t * 16 + r1);
            async_copy_b128(bb + (unsigned)(r0 * VS_PITCH * 4) + (unsigned)(ln * 16),
                            (unsigned long long)(uintptr_t)sA + (unsigned)(ln * 16));
            async_copy_b128(bb + (unsigned)(r1 * VS_PITCH * 4) + (unsigned)(ln * 16),
                            (unsigned long long)(uintptr_t)sB + (unsigned)(ln * 16));
        };

        DRAIN_DS();
        issueV(0, 0);
        for (int t = 0; t < NT; ++t) {
            if (t + 1 < NT) {
                DRAIN_DS();               // own frag reads of target buffer done
                issueV(t + 1, (t + 1) & 1);
                WAIT_ASYNC(2);
            } else {
                WAIT_ASYNC(0);
            }
            __syncthreads();                      // all waves' tile-t data visible
            const float* vs = &smem[KST_OFF + (t & 1) * VS_BUF];
            const int sb = t * 16;
            for (int s0 = 0; s0 < 16; s0 += 4) {
                // A = P[16(g-pad) x 4(s)]: rows >= G are zero
                const int arow = (lo < G) ? lo : 0;
                v2f a = *(const v2f*)(&smem[SC_OFF + arow * SC_PITCH + sb + s0 + hoff]);
                if (lo >= G) { a.x = 0.f; a.y = 0.f; }
                // B = V[4(s) x 16(d-cols)]; wave w owns d-cols 16w..16w+15
                const int srow = s0 + hoff;
                v2f bb;
                bb.x = vs[(srow    ) * VS_PITCH + wv * 16 + lo];
                bb.y = vs[(srow + 1) * VS_PITCH + wv * 16 + lo];
                acc = wmma_f32(a, bb, acc);
            }
            __syncthreads();                      // protect buffer before re-stage
        }
    }

    // out[b, kh*G + g, d] ; D layout: vgpr r = g, lanes 0-15 = d columns
    if (!hiH) {
        const size_t obase = ((size_t)(b * KH + kh) * G) * D;
        const int dcol = wv * 16 + lo;
        out[obase + 0 * D + dcol] = acc[0] * smem[STAT_OFF + 4 + 0];
        out[obase + 1 * D + dcol] = acc[1] * smem[STAT_OFF + 4 + 1];
        out[obase + 2 * D + dcol] = acc[2] * smem[STAT_OFF + 4 + 2];
        out[obase + 3 * D + dcol] = acc[3] * smem[STAT_OFF + 4 + 3];
    }
}

extern "C" void kernel_launch(void* const* d_in, const int* in_sizes, int n_in,
                              void* d_out, int out_size, void* d_ws, size_t ws_size,
                              hipStream_t stream) {
    const float* q    = (const float*)d_in[0];
    const float* k    = (const float*)d_in[1];
    const float* v    = (const float*)d_in[2];
    const float* kbuf = (const float*)d_in[3];
    const float* vbuf = (const float*)d_in[4];
    const int* rtt    = (const int*)d_in[5];
    const int* rpi    = (const int*)d_in[6];
    const int* ocl    = (const int*)d_in[7];
    const int* slen   = (const int*)d_in[8];
    float* out = (float*)d_out;

    dim3 grid(B * KH);   // 256 workgroups: one per (request, kv-head)
    dim3 block(256);     // 8 wave32s
    decode_gqa_wmma_f32<<<grid, block, 0, stream>>>(
        q, k, v, kbuf, vbuf, rtt, rpi, ocl, slen, out);
}